// DetectionPostProcessor_49228915147541
// MI455X (gfx1250) — compile-verified
//
#include <hip/hip_runtime.h>
#include <hip/hip_bf16.h>
#include <stdint.h>

// Detection post-process for MI455X (gfx1250), compile-only tuned.
// Memory-bound: ~86 MB effective traffic -> ~3.7 us floor at 23.3 TB/s.
// No matmul structure -> WMMA not applicable; CDNA5 paths used instead:
//   * global_load_async_to_lds_b128 (ASYNCcnt-tracked async copy to LDS)
//   * s_wait_asynccnt split counter wait
//   * wave32 shuffle argmax reductions (2 barriers/iter instead of 11)

#define Bn 8
#define Cn 10
#define Hh 512
#define Ww 512
#define Kk 50
#define THRv 0.3f
#define TILE_H 16
#define NBLK (Hh / TILE_H)        // 32 row-tiles per plane
#define PLANES (Bn * Cn)          // 80
#define PLANE_SZ (Hh * Ww)        // 262144
#define CAND (NBLK * Kk)          // 1600 candidates per plane
#define NEG_BIG (-1.0e30f)

__device__ __forceinline__ void async_load_b128_to_lds(uint32_t lds_byte_off,
                                                       const float* gaddr) {
  // CDNA5 async path: per-lane 16B global -> LDS, tracked by ASYNCcnt.
  asm volatile("global_load_async_to_lds_b128 %0, %1, off"
               :: "v"(lds_byte_off), "v"(gaddr)
               : "memory");
}

__device__ __forceinline__ void wait_asynccnt0() {
  asm volatile("s_wait_asynccnt 0" ::: "memory");
}

// wave32 argmax (tie-break: smaller index). 5 shuffle steps, no barriers.
__device__ __forceinline__ void wave_argmax2(float& v, int& i) {
#pragma unroll
  for (int off = 16; off > 0; off >>= 1) {
    const float ov = __shfl_xor(v, off, 32);
    const int   oi = __shfl_xor(i, off, 32);
    if (ov > v || (ov == v && oi < i)) { v = ov; i = oi; }
  }
}

__device__ __forceinline__ void wave_argmax3(float& v, int& i, int& s) {
#pragma unroll
  for (int off = 16; off > 0; off >>= 1) {
    const float ov = __shfl_xor(v, off, 32);
    const int   oi = __shfl_xor(i, off, 32);
    const int   os = __shfl_xor(s, off, 32);
    if (ov > v || (ov == v && oi < i)) { v = ov; i = oi; s = os; }
  }
}

// ---------------------------------------------------------------------------
// Kernel 1: fused 3x3 max-pool NMS + threshold + block-local top-50.
// grid = (NBLK, PLANES), block = 512 threads (16 wave32).
// ---------------------------------------------------------------------------
__global__ __launch_bounds__(512)
void peaks_blocktopk_kernel(const float* __restrict__ cls,
                            float* __restrict__ ws_s,
                            int* __restrict__ ws_i) {
  const int tile  = blockIdx.x;          // 0..31  (row tile)
  const int plane = blockIdx.y;          // 0..79  (b*C + c)
  const int tid   = threadIdx.x;         // 0..511 == column
  const int lane  = tid & 31;
  const int wave  = tid >> 5;            // 0..15
  const int r0    = tile * TILE_H;

  __shared__ __align__(16) float smem[(TILE_H + 2) * Ww];  // 36 KB halo tile
  __shared__ float red_s[16];
  __shared__ int   red_i[16];

  const float* pbase = cls + (size_t)plane * PLANE_SZ;
  const uint32_t smem_base = (uint32_t)(uintptr_t)&smem[0];

  // ---- Stage tile rows [r0-1, r0+TILE_H] into LDS via async copy ----------
  const int nvec = (TILE_H + 2) * (Ww / 4);   // 18 * 128 = 2304 vec4 chunks
  for (int i = tid; i < nvec; i += 512) {
    const int trow = i >> 7;                  // tile row 0..17
    const int c4   = i & 127;                 // vec4 column
    const int grow = r0 - 1 + trow;           // global row (may be OOB)
    const uint32_t loff = smem_base + (uint32_t)((trow * Ww + c4 * 4) * 4);
    if (0 <= grow && grow < Hh) {
      async_load_b128_to_lds(loff, pbase + (size_t)grow * Ww + c4 * 4);
    } else {
      float* lp = &smem[trow * Ww + c4 * 4];  // -inf pad rows (SAME boundary)
      lp[0] = NEG_BIG; lp[1] = NEG_BIG; lp[2] = NEG_BIG; lp[3] = NEG_BIG;
    }
  }
  wait_asynccnt0();
  __syncthreads();

  // ---- Per-thread peak test for its column (fully unrolled -> registers) --
  const int col = tid;
  float pv[TILE_H];
  float cmax = 0.0f;
  int   cj   = 0;
#pragma unroll
  for (int j = 0; j < TILE_H; ++j) {
    const int tr = j + 1;                     // tile row of center
    const float s = smem[tr * Ww + col];
    float m = s;
#pragma unroll
    for (int dr = -1; dr <= 1; ++dr) {
      const int rr = (tr + dr) * Ww;
      const float lv = (col > 0)      ? smem[rr + col - 1] : NEG_BIG;
      const float cv =                   smem[rr + col];
      const float rv = (col < Ww - 1) ? smem[rr + col + 1] : NEG_BIG;
      m = fmaxf(m, fmaxf(lv, fmaxf(cv, rv)));
    }
    const float p = (s == m && s > THRv) ? s : 0.0f;
    pv[j] = p;
    if (p > cmax) { cmax = p; cj = j; }
  }

  // ---- Block top-50: wave32 shuffle argmax + tiny cross-wave combine ------
  uint32_t removed = 0;
  const int out_base = (plane * NBLK + tile) * Kk;
  for (int k = 0; k < Kk; ++k) {
    float v = cmax;
    int   ix = ((r0 + cj) << 9) | col;        // flat plane index (W = 2^9)
    wave_argmax2(v, ix);                      // 5 shuffle steps, no barrier
    if (lane == 0) { red_s[wave] = v; red_i[wave] = ix; }
    __syncthreads();
    if (wave == 0) {                          // first wave combines 16 waves
      float v2 = (lane < 16) ? red_s[lane] : 0.0f;
      int   i2 = (lane < 16) ? red_i[lane] : (1 << 30);
      wave_argmax2(v2, i2);
      if (lane == 0) {
        red_s[0] = v2; red_i[0] = i2;
        ws_s[out_base + k] = v2;
        ws_i[out_base + k] = i2;
      }
    }
    __syncthreads();
    const float wsc = red_s[0];
    const int   wix = red_i[0];
    // Winner thread removes its element and rescans its register array.
    if (wsc > 0.0f && (wix & 511) == col) {
      removed |= 1u << ((wix >> 9) - r0);
      cmax = 0.0f; cj = 0;
#pragma unroll
      for (int j = 0; j < TILE_H; ++j) {
        const float p = ((removed >> j) & 1u) ? 0.0f : pv[j];
        if (p > cmax) { cmax = p; cj = j; }
      }
    }
  }
}

// ---------------------------------------------------------------------------
// Kernel 2: per-plane merge of 1600 candidates -> top-50, box decode + write.
// grid = PLANES, block = 256 threads (8 wave32).
// ---------------------------------------------------------------------------
__global__ __launch_bounds__(256)
void merge_decode_kernel(const float* __restrict__ bbox,
                         const float* __restrict__ ws_s,
                         const int* __restrict__ ws_i,
                         float* __restrict__ out) {
  const int plane = blockIdx.x;               // b*C + c
  const int tid   = threadIdx.x;
  const int lane  = tid & 31;
  const int wave  = tid >> 5;                 // 0..7
  const int b = plane / Cn;
  const int c = plane % Cn;

  __shared__ float cs[CAND];
  __shared__ int   ci[CAND];
  __shared__ float sel_s[Kk];
  __shared__ int   sel_i[Kk];
  __shared__ float red_s[8];
  __shared__ int   red_i[8];
  __shared__ int   red_k[8];

  for (int i = tid; i < CAND; i += 256) {
    cs[i] = ws_s[plane * CAND + i];
    ci[i] = ws_i[plane * CAND + i];
  }
  __syncthreads();

  for (int k = 0; k < Kk; ++k) {
    // Each thread scans its ~7 strided slots.
    float m = 0.0f; int mi = 1 << 30; int ms = -1;
    for (int i = tid; i < CAND; i += 256) {
      const float v = cs[i];
      const int   ix = ci[i];
      if (v > m || (v == m && v > 0.0f && ix < mi)) { m = v; mi = ix; ms = i; }
    }
    wave_argmax3(m, mi, ms);                  // intra-wave, no barrier
    if (lane == 0) { red_s[wave] = m; red_i[wave] = mi; red_k[wave] = ms; }
    __syncthreads();
    if (wave == 0) {                          // combine 8 wave winners
      float v2 = (lane < 8) ? red_s[lane] : 0.0f;
      int   i2 = (lane < 8) ? red_i[lane] : (1 << 30);
      int   s2 = (lane < 8) ? red_k[lane] : -1;
      wave_argmax3(v2, i2, s2);
      if (lane == 0) {
        sel_s[k] = v2;
        sel_i[k] = (v2 > 0.0f) ? i2 : 0;
        if (v2 > 0.0f && s2 >= 0) cs[s2] = 0.0f;   // remove winner
      }
    }
    __syncthreads();
  }

  // ---- Decode + masked writeout (lanes 0..49) ----------------------------
  if (tid < Kk) {
    const int k = tid;
    const float s = sel_s[k];
    const int idx = sel_i[k];
    const float vf = (s > THRv) ? 1.0f : 0.0f;
    const int hi = idx >> 9;
    const int wi = idx & 511;

    const float* pb = bbox + (size_t)b * 7 * PLANE_SZ + idx;  // sparse gather
    const float p0 = pb[0 * PLANE_SZ];
    const float p1 = pb[1 * PLANE_SZ];
    const float p2 = pb[2 * PLANE_SZ];
    const float p3 = pb[3 * PLANE_SZ];
    const float p4 = pb[4 * PLANE_SZ];
    const float p5 = pb[5 * PLANE_SZ];
    const float p6 = pb[6 * PLANE_SZ];

    const float res = 102.4f / 512.0f;       // 0.2
    const float x   = -51.2f + ((float)wi + 0.5f) * res + p0;
    const float y   = -51.2f + ((float)hi + 0.5f) * res + p1;
    const float z   = p2;
    const float w   = expf(fminf(p3, 10.0f));
    const float l   = expf(fminf(p4, 10.0f));
    const float hh  = expf(fminf(p5, 10.0f));
    const float yaw = p6;

    const int base = plane * Kk + k;          // (b*C + c)*K + k
    float* boxes = out;                       // 28000 floats
    boxes[base * 7 + 0] = x   * vf;
    boxes[base * 7 + 1] = y   * vf;
    boxes[base * 7 + 2] = z   * vf;
    boxes[base * 7 + 3] = w   * vf;
    boxes[base * 7 + 4] = l   * vf;
    boxes[base * 7 + 5] = hh  * vf;
    boxes[base * 7 + 6] = yaw * vf;
    out[28000 + base] = s * vf;               // scores
    out[32000 + base] = (float)c;             // labels
    out[36000 + base] = (float)b;             // batch_idx
    out[40000 + base] = vf;                   // valid
  }
}

extern "C" void kernel_launch(void* const* d_in, const int* in_sizes, int n_in,
                              void* d_out, int out_size, void* d_ws, size_t ws_size,
                              hipStream_t stream) {
  (void)in_sizes; (void)n_in; (void)out_size; (void)ws_size;
  const float* cls  = (const float*)d_in[0];   // (8,10,512,512) f32
  const float* bbox = (const float*)d_in[1];   // (8,7,512,512)  f32
  float* out = (float*)d_out;                  // 44000 f32 (concat outputs)

  // Workspace: per-block candidate lists: 80*32*50 scores + 80*32*50 indices.
  float* ws_s = (float*)d_ws;
  int*   ws_i = (int*)((char*)d_ws + (size_t)PLANES * CAND * sizeof(float));

  dim3 g1(NBLK, PLANES);                       // 32 x 80 blocks
  peaks_blocktopk_kernel<<<g1, 512, 0, stream>>>(cls, ws_s, ws_i);
  merge_decode_kernel<<<PLANES, 256, 0, stream>>>(bbox, ws_s, ws_i, out);
}